// DeepSNF_14018773254826
// MI455X (gfx1250) — compile-verified
//
#include <hip/hip_runtime.h>

// ---- problem constants (match reference) ----
#define MMOD   3
#define INSZ   50000
#define HEADS  4
#define DIMH   64
#define HID    256
#define EMB    512
#define N0C    49152
#define N1C    16384
#define BN     8192
#define E0C    262144
#define E1C    131072

#define KC     32      // K-chunk staged in LDS per block
#define LDSTR  34      // LDS column stride ([n][kk] layout, padded 32->34)

typedef __attribute__((ext_vector_type(2))) float v2f;
typedef __attribute__((ext_vector_type(8))) float v8f;

// =====================================================================
// Generic f32 WMMA GEMM:  C[M,N] = op(A)[M,K] * op(B)[K,N] (+ biases)
//   GATHER_A : A row m comes from A[aidx[m]*lda + k]
//   ABIAS    : add abias[k] to each A element (fused pre_b)
//   TRANS_B  : B element (k,n) read from Bm[n*ldb + k]  (for emb @ emb.T)
//   OBIAS    : add obias[n] to output columns
//   NT_C     : nontemporal (streaming) C stores — keep L2 for B operand
// Block = 256 threads (8 waves). Block tile M=128, N=64. B tile staged
// in LDS as [n][kk] (64 x KC, stride 34) so each WMMA B fragment is one
// aligned ds_load_b64. V_WMMA_F32_16X16X4_F32 layouts:
//   A 16x4 : lane&15 = row M, (lane>>4)*2+v = K      (2 VGPRs)
//   B 4x16 : lane&15 = col N, (lane>>4)*2+v = K      (2 VGPRs)
//   C 16x16: lane&15 = col N, row M = v + 8*(lane>>4) (8 VGPRs)
// =====================================================================
template<bool GATHER_A, bool ABIAS, bool TRANS_B, bool OBIAS, bool NT_C>
__global__ __launch_bounds__(256)
void wmma_gemm_f32(const float* __restrict__ A, const int* __restrict__ aidx,
                   const float* __restrict__ abias, const float* __restrict__ Bm,
                   const float* __restrict__ obias, float* __restrict__ C,
                   int Mrows, int Ncols, int K, int lda, int ldb, int ldc)
{
    (void)Mrows; (void)Ncols;
    __shared__ float Bs[64 * LDSTR];      // [n][kk], kk-pairs contiguous

    const int tid  = threadIdx.x;
    const int lane = tid & 31;
    const int wave = tid >> 5;
    const int r    = lane & 15;
    const int grp  = lane >> 4;           // 0 or 1 (K split)
    const int row_base = blockIdx.y * 128 + wave * 16;
    const int col_base = blockIdx.x * 64;

    long arow = row_base + r;
    if (GATHER_A) arow = aidx[arow];
    const float* __restrict__ pa = A + arow * (long)lda;

    v8f cacc[4] = { v8f{0.f,0.f,0.f,0.f,0.f,0.f,0.f,0.f},
                    v8f{0.f,0.f,0.f,0.f,0.f,0.f,0.f,0.f},
                    v8f{0.f,0.f,0.f,0.f,0.f,0.f,0.f,0.f},
                    v8f{0.f,0.f,0.f,0.f,0.f,0.f,0.f,0.f} };

    for (int k0 = 0; k0 < K; k0 += KC) {
        // ---- cooperative load of B tile (KC x 64) into LDS in [n][kk] layout ----
#pragma unroll
        for (int rload = 0; rload < (KC * 64) / (256 * 2); ++rload) {
            const int e = rload * 256 + tid;          // e in [0, 1024): v2f units
            if (TRANS_B) {
                // contiguous along k on both sides: one b64 load + one b64 store
                const int n   = e >> 4;
                const int kkp = (e & 15) * 2;
                v2f v = *(const v2f*)(&Bm[(long)(col_base + n) * ldb + (k0 + kkp)]);
                *(v2f*)(&Bs[n * LDSTR + kkp]) = v;
            } else {
                // coalesced along n in global; scatter as two b32 LDS stores
                const int kk = e >> 5;
                const int np = (e & 31) * 2;
                v2f v = *(const v2f*)(&Bm[(long)(k0 + kk) * ldb + (col_base + np)]);
                Bs[np * LDSTR + kk]       = v.x;
                Bs[(np + 1) * LDSTR + kk] = v.y;
            }
        }
        __syncthreads();

        // ---- compute over the K-chunk ----
#pragma unroll
        for (int kk = 0; kk < KC; kk += 4) {
            const int ka = kk + grp * 2;
            v2f a = *(const v2f*)(pa + k0 + ka);          // global_load_b64
            if (ABIAS) {
                v2f ab = *(const v2f*)(abias + k0 + ka);
                a.x += ab.x; a.y += ab.y;
            }
#pragma unroll
            for (int j = 0; j < 4; ++j) {
                const int n = j * 16 + r;
                v2f b = *(const v2f*)(&Bs[n * LDSTR + ka]);   // ds_load_b64
                cacc[j] = __builtin_amdgcn_wmma_f32_16x16x4_f32(
                    false, a, false, b, (short)0, cacc[j], false, false);
            }
        }
        __syncthreads();
    }

#pragma unroll
    for (int j = 0; j < 4; ++j) {
        const int col = col_base + j * 16 + r;
        const float ob = OBIAS ? obias[col] : 0.0f;
#pragma unroll
        for (int v = 0; v < 8; ++v) {
            const int row = row_base + v + 8 * grp;
            float val = cacc[j][v] + ob;
            if (NT_C) __builtin_nontemporal_store(val, &C[(long)row * ldc + col]);
            else      C[(long)row * ldc + col] = val;
        }
    }
}

// =====================================================================
// Elementwise / scatter helpers
// =====================================================================
__global__ void init_f32(float* __restrict__ p, float v, int n) {
    int i = blockIdx.x * blockDim.x + threadIdx.x;
    if (i < n) p[i] = v;
}

// per-node, per-head attention score: s[n,h] = sum_d h[n, h*64+d] * att[h,d]
__global__ void score_kernel(const float* __restrict__ h, const float* __restrict__ att,
                             float* __restrict__ s, int nNodes) {
    int idx = blockIdx.x * blockDim.x + threadIdx.x;
    if (idx >= nNodes * HEADS) return;
    int node = idx >> 2, hd = idx & 3;
    const float* hp = h + (long)node * HID + hd * DIMH;
    const float* ap = att + hd * DIMH;
    float acc = 0.f;
#pragma unroll 4
    for (int d = 0; d < DIMH; ++d) acc += hp[d] * ap[d];
    s[idx] = acc;
}

__device__ inline float leaky02(float x) { return x > 0.f ? x : 0.2f * x; }

__device__ inline void atomicMaxF(float* addr, float val) {
    if (val >= 0.f) atomicMax((int*)addr, __float_as_int(val));
    else            atomicMin((unsigned int*)addr, __float_as_uint(val));
}

__global__ void attn_max_kernel(const float* __restrict__ s, const int* __restrict__ src,
                                const int* __restrict__ dst, float* __restrict__ mx, int nE) {
    int idx = blockIdx.x * blockDim.x + threadIdx.x;
    if (idx >= nE * HEADS) return;
    int e = idx >> 2, hh = idx & 3;
    float lg = leaky02(s[src[e] * HEADS + hh]);
    atomicMaxF(&mx[dst[e] * HEADS + hh], lg);
}

__global__ void attn_den_kernel(const float* __restrict__ s, const int* __restrict__ src,
                                const int* __restrict__ dst, const float* __restrict__ mx,
                                float* __restrict__ den, int nE) {
    int idx = blockIdx.x * blockDim.x + threadIdx.x;
    if (idx >= nE * HEADS) return;
    int e = idx >> 2, hh = idx & 3;
    float lg = leaky02(s[src[e] * HEADS + hh]);
    float ex = expf(lg - mx[dst[e] * HEADS + hh]);
    atomicAdd(&den[dst[e] * HEADS + hh], ex);
}

// one block per edge; 256 threads = 256 channels
__global__ __launch_bounds__(256)
void msg_kernel(const float* __restrict__ h, const float* __restrict__ s,
                const float* __restrict__ mx, const float* __restrict__ den,
                const int* __restrict__ src, const int* __restrict__ dst,
                const float* __restrict__ vals, float* __restrict__ xout) {
    int e  = blockIdx.x;
    int c  = threadIdx.x;
    int hh = c >> 6;
    int sN = src[e], dN = dst[e];
    float lg    = leaky02(s[sN * HEADS + hh]);
    float ex    = expf(lg - mx[dN * HEADS + hh]);
    float alpha = ex / (den[dN * HEADS + hh] + 1e-16f);
    float m     = h[(long)sN * HID + c] * alpha * vals[e];
    atomicAdd(&xout[(long)dN * HID + c], m);
}

__global__ void add_bias_kernel(float* __restrict__ x, const float* __restrict__ bvec, int n) {
    int i = blockIdx.x * blockDim.x + threadIdx.x;
    if (i < n) x[i] += bvec[i & (HID - 1)];
}

// interp masks (evaluate=True branch of reference)
__global__ void interp_kernel(const float* __restrict__ masks, float* __restrict__ imask) {
    int b = blockIdx.x * blockDim.x + threadIdx.x;
    if (b >= BN) return;
    float m0 = masks[b * 3 + 0], m1 = masks[b * 3 + 1], m2 = masks[b * 3 + 2];
    float ms  = m0 + m1 + m2;
    float add = powf(4.0f, -20.0f) + powf(ms, -20.0f);   // rm.sum(-1)==3 -> 1/(1+3)^20
    float rm  = floorf(1.0f + add);
    rm = rm / (rm + 1e-10f);
    float z0 = m0 * rm; z0 = z0 + (1.0f - z0) * (-1e10f);
    float z1 = m1 * rm; z1 = z1 + (1.0f - z1) * (-1e10f);
    float z2 = m2 * rm; z2 = z2 + (1.0f - z2) * (-1e10f);
    float zm = fmaxf(z0, fmaxf(z1, z2));
    float e0 = expf(z0 - zm), e1 = expf(z1 - zm), e2 = expf(z2 - zm);
    float inv = 1.0f / (e0 + e1 + e2);
    imask[b * 3 + 0] = e0 * inv;
    imask[b * 3 + 1] = e1 * inv;
    imask[b * 3 + 2] = e2 * inv;
}

__global__ void weights_kernel(const float* __restrict__ iw, float* __restrict__ wts,
                               float* __restrict__ w_out) {
    float a = iw[0], b = iw[1], c = iw[2];
    float mx = fmaxf(a, fmaxf(b, c));
    float e0 = expf(a - mx), e1 = expf(b - mx), e2 = expf(c - mx);
    float inv = 1.0f / (e0 + e1 + e2);
    wts[0] = e0 * inv; wts[1] = e1 * inv; wts[2] = e2 * inv;
    w_out[0] = wts[0]; w_out[1] = wts[1]; w_out[2] = wts[2];
}

// acc += weights[i] * imask[:,i] * (x1[res_n_id] + x2)
__global__ void acc_kernel(const float* __restrict__ x1, const float* __restrict__ x2,
                           const int* __restrict__ res, const float* __restrict__ imask,
                           const float* __restrict__ wts, int i, float* __restrict__ acc) {
    int idx = blockIdx.x * blockDim.x + threadIdx.x;   // BN*HID
    if (idx >= BN * HID) return;
    int b = idx >> 8, c = idx & (HID - 1);
    float xpre = x1[(long)res[b] * HID + c];
    float v    = xpre + x2[idx];
    acc[idx] += wts[i] * imask[b * 3 + i] * v;
}

// =====================================================================
extern "C" void kernel_launch(void* const* d_in, const int* in_sizes, int n_in,
                              void* d_out, int out_size, void* d_ws, size_t ws_size,
                              hipStream_t stream) {
    const int*   n_id     = (const int*)  d_in[0];
    const int*   src0     = (const int*)  d_in[1];
    const int*   dst0     = (const int*)  d_in[2];
    const int*   src1     = (const int*)  d_in[3];
    const int*   dst1     = (const int*)  d_in[4];
    const int*   res_n_id = (const int*)  d_in[5];
    const float* vals0    = (const float*)d_in[6];
    const float* vals1    = (const float*)d_in[7];
    const float* masks    = (const float*)d_in[8];
    const float* pre_W    = (const float*)d_in[9];
    const float* pre_b    = (const float*)d_in[10];
    const float* W_gat    = (const float*)d_in[11];
    const float* att      = (const float*)d_in[12];
    const float* gat_b    = (const float*)d_in[13];
    const float* interp_w = (const float*)d_in[14];
    const float* emb_W    = (const float*)d_in[15];
    const float* emb_b    = (const float*)d_in[16];

    float* dot_out = (float*)d_out;                       // [BN, BN]
    float* emb_out = dot_out + (long)BN * BN;             // [BN, EMB]
    float* w_out   = emb_out + (long)BN * EMB;            // [3]

    // ---- workspace carve-up ----
    char*  ws  = (char*)d_ws;
    size_t off = 0;
    auto carve = [&](size_t nfloats) -> float* {
        float* p = (float*)(ws + off);
        off += nfloats * sizeof(float);
        off = (off + 255) & ~(size_t)255;
        return p;
    };
    float* h0    = carve((size_t)N0C * HID);
    float* s0    = carve((size_t)N0C * HEADS);
    float* x1    = carve((size_t)N1C * HID);
    float* h1    = carve((size_t)N1C * HID);
    float* s1    = carve((size_t)N1C * HEADS);
    float* mx0   = carve((size_t)N1C * HEADS);
    float* den0  = carve((size_t)N1C * HEADS);
    float* x2    = carve((size_t)BN * HID);
    float* mx1   = carve((size_t)BN * HEADS);
    float* den1  = carve((size_t)BN * HEADS);
    float* accb  = carve((size_t)BN * HID);
    float* imask = carve((size_t)BN * 3);
    float* wts   = carve(4);

    const dim3 blk(256);
    auto nblk = [](int n) { return (unsigned)((n + 255) / 256); };

    // interpolation weights & masks
    weights_kernel<<<1, 1, 0, stream>>>(interp_w, wts, w_out);
    interp_kernel<<<nblk(BN), blk, 0, stream>>>(masks, imask);
    init_f32<<<nblk(BN * HID), blk, 0, stream>>>(accb, 0.f, BN * HID);

    for (int i = 0; i < MMOD; ++i) {
        const float* preW_i = pre_W + (long)i * INSZ * HID;
        const float* preb_i = pre_b + (long)i * HID;
        const float* Wg_i   = W_gat + (long)i * HID * HID;
        const float* att_i  = att   + (long)i * HEADS * DIMH;
        const float* gb_i   = gat_b + (long)i * HID;
        const int*   nid_i  = n_id  + (long)i * N0C;
        const int*   s0_i   = src0  + (long)i * E0C;
        const int*   d0_i   = dst0  + (long)i * E0C;
        const int*   s1_i   = src1  + (long)i * E1C;
        const int*   d1_i   = dst1  + (long)i * E1C;
        const int*   res_i  = res_n_id + (long)i * BN;
        const float* v0_i   = vals0 + (long)i * E0C;
        const float* v1_i   = vals1 + (long)i * E1C;

        // ---- layer 0: h0 = (pre_W[n_id] + pre_b) @ Wg  [N0,256]x[256,256]
        wmma_gemm_f32<true, true, false, false, false>
            <<<dim3(HID / 64, N0C / 128), blk, 0, stream>>>(
                preW_i, nid_i, preb_i, Wg_i, nullptr, h0,
                N0C, HID, HID, HID, HID, HID);
        score_kernel<<<nblk(N0C * HEADS), blk, 0, stream>>>(h0, att_i, s0, N0C);

        init_f32<<<nblk(N1C * HEADS), blk, 0, stream>>>(mx0, -3.0e38f, N1C * HEADS);
        init_f32<<<nblk(N1C * HEADS), blk, 0, stream>>>(den0, 0.f, N1C * HEADS);
        init_f32<<<nblk(N1C * HID),   blk, 0, stream>>>(x1, 0.f, N1C * HID);

        attn_max_kernel<<<nblk(E0C * HEADS), blk, 0, stream>>>(s0, s0_i, d0_i, mx0, E0C);
        attn_den_kernel<<<nblk(E0C * HEADS), blk, 0, stream>>>(s0, s0_i, d0_i, mx0, den0, E0C);
        msg_kernel<<<E0C, blk, 0, stream>>>(h0, s0, mx0, den0, s0_i, d0_i, v0_i, x1);
        add_bias_kernel<<<nblk(N1C * HID), blk, 0, stream>>>(x1, gb_i, N1C * HID);

        // ---- layer 1: h1 = x1 @ Wg  [N1,256]x[256,256]
        wmma_gemm_f32<false, false, false, false, false>
            <<<dim3(HID / 64, N1C / 128), blk, 0, stream>>>(
                x1, nullptr, nullptr, Wg_i, nullptr, h1,
                N1C, HID, HID, HID, HID, HID);
        score_kernel<<<nblk(N1C * HEADS), blk, 0, stream>>>(h1, att_i, s1, N1C);

        init_f32<<<nblk(BN * HEADS), blk, 0, stream>>>(mx1, -3.0e38f, BN * HEADS);
        init_f32<<<nblk(BN * HEADS), blk, 0, stream>>>(den1, 0.f, BN * HEADS);
        init_f32<<<nblk(BN * HID),   blk, 0, stream>>>(x2, 0.f, BN * HID);

        attn_max_kernel<<<nblk(E1C * HEADS), blk, 0, stream>>>(s1, s1_i, d1_i, mx1, E1C);
        attn_den_kernel<<<nblk(E1C * HEADS), blk, 0, stream>>>(s1, s1_i, d1_i, mx1, den1, E1C);
        msg_kernel<<<E1C, blk, 0, stream>>>(h1, s1, mx1, den1, s1_i, d1_i, v1_i, x2);
        add_bias_kernel<<<nblk(BN * HID), blk, 0, stream>>>(x2, gb_i, BN * HID);

        // ---- acc += w_i * imask_i * (x1[res_n_id] + x2)
        acc_kernel<<<nblk(BN * HID), blk, 0, stream>>>(x1, x2, res_i, imask, wts, i, accb);
    }

    // ---- emb = acc @ emb_W + emb_b   [8192,256]x[256,512]
    wmma_gemm_f32<false, false, false, true, false>
        <<<dim3(EMB / 64, BN / 128), blk, 0, stream>>>(
            accb, nullptr, nullptr, emb_W, emb_b, emb_out,
            BN, EMB, HID, HID, EMB, EMB);

    // ---- dot = emb @ emb.T   [8192,512]x[512,8192]  (B transposed, NT stores)
    wmma_gemm_f32<false, false, true, false, true>
        <<<dim3(BN / 64, BN / 128), blk, 0, stream>>>(
            emb_out, nullptr, nullptr, emb_out, nullptr, dot_out,
            BN, BN, EMB, EMB, EMB, BN);
}